// TopKSAE_24343874633902
// MI455X (gfx1250) — compile-verified
//
#include <hip/hip_runtime.h>

typedef __attribute__((ext_vector_type(2))) float v2f;
typedef __attribute__((ext_vector_type(8))) float v8f;

#define D_MODEL 1024
#define D_SAE   32768
#define TOPK    64
#define BATCH   4096

// --------------------------------------------------------------------------
// Encoder GEMM: z = (x - b_dec) @ W_enc + b_enc
// Workgroup tile 64(M) x 128(N), K-chunk 32, 8 waves.
// Each wave computes a 16x64 strip as 4 WMMA 16x16 subtiles (f32 16x16x4).
// LDS layouts are fragment-native so every WMMA operand is one ds_load_b64:
//   A: row-major, even stride 34  -> (k,k+1) contiguous, 8B aligned
//   B: K-quad interleaved [k/4][n][4] -> per-lane (k,k+1)/(k+2,k+3) pair
//      contiguous; lanes 0-15 hit banks 4n..4n+1, lanes 16-31 4n+2..4n+3
//      (all 64 banks, conflict-free).
// --------------------------------------------------------------------------
__global__ __launch_bounds__(256) void sae_encode(const float* __restrict__ x,
                                                  const float* __restrict__ W_enc,
                                                  const float* __restrict__ b_enc,
                                                  const float* __restrict__ b_dec,
                                                  float* __restrict__ z)
{
    __shared__ float As[64 * 34];   // 8704 B
    __shared__ float Bs[8 * 512];   // 16384 B  (8 K-quads x 128 N x 4)

    const int tid  = threadIdx.x;
    const int lane = tid & 31;
    const int wid  = tid >> 5;
    const int wm   = (wid & 3) << 4;   // wave M offset: 0,16,32,48
    const int wn   = (wid >> 2) << 6;  // wave N offset: 0,64
    const int row0 = blockIdx.y * 64;
    const int col0 = blockIdx.x * 128;

    v8f acc[4] = {};

    for (int k0 = 0; k0 < D_MODEL; k0 += 32) {
        // stage A = x - b_dec  (64x32)
        #pragma unroll
        for (int i = 0; i < 8; ++i) {
            int e = tid + i * 256;
            int m = e >> 5, k = e & 31;
            As[m * 34 + k] = x[(size_t)(row0 + m) * D_MODEL + k0 + k] - b_dec[k0 + k];
        }
        // stage B = W_enc chunk (32x128), K-quad interleaved
        #pragma unroll
        for (int i = 0; i < 16; ++i) {
            int e = tid + i * 256;
            int k = e >> 7, n = e & 127;
            Bs[(k >> 2) * 512 + (n << 2) + (k & 3)] =
                W_enc[(size_t)(k0 + k) * D_SAE + col0 + n];
        }

        // prefetch next K-chunk into cache while this chunk computes
        // (global_prefetch_b8; covers 16 KB of W_enc + 8 KB of x)
        if (k0 + 32 < D_MODEL) {
            if (tid < 128) {
                int pk = tid >> 2, seg = tid & 3;
                __builtin_prefetch(
                    &W_enc[(size_t)(k0 + 32 + pk) * D_SAE + col0 + seg * 32], 0, 1);
            }
            if (tid < 64)
                __builtin_prefetch(
                    &x[(size_t)(row0 + tid) * D_MODEL + k0 + 32], 0, 1);
        }
        __syncthreads();

        const int la = lane & 15;
        const int hi = lane >> 4;            // 0: K pair (k,k+1); 1: (k+2,k+3)
        #pragma unroll
        for (int kk = 0; kk < 32; kk += 4) {
            const int ka = kk + (hi << 1);
            v2f a = *(const v2f*)&As[(wm + la) * 34 + ka];
            #pragma unroll
            for (int ns = 0; ns < 4; ++ns) {
                const int nb = wn + (ns << 4) + la;
                v2f b = *(const v2f*)&Bs[(kk >> 2) * 512 + (nb << 2) + (hi << 1)];
                acc[ns] = __builtin_amdgcn_wmma_f32_16x16x4_f32(
                    false, a, false, b, (short)0, acc[ns], false, false);
            }
        }
        __syncthreads();
    }

    // epilogue: C layout -> VGPR r holds M=r (lanes 0-15) / M=r+8 (lanes 16-31)
    const int la = lane & 15;
    const int mh = (lane >> 4) << 3;
    #pragma unroll
    for (int ns = 0; ns < 4; ++ns) {
        const int col = col0 + wn + (ns << 4) + la;
        const float be = b_enc[col];
        #pragma unroll
        for (int r = 0; r < 8; ++r) {
            const int row = row0 + wm + mh + r;
            z[(size_t)row * D_SAE + col] = acc[ns][r] + be;
        }
    }
}

// --------------------------------------------------------------------------
// Per-row relu + top-64 select via 2-pass radix on 16-bit float keys kept in
// LDS (64 KB of the 320 KB WGP LDS), exact tie resolution, bitonic sort of
// the final 64, scatter into dense f and emit indices. Also zeroes f row.
// --------------------------------------------------------------------------
__global__ __launch_bounds__(256) void sae_topk(const float* __restrict__ z,
                                                float* __restrict__ f,
                                                float* __restrict__ idx_out)
{
    __shared__ unsigned short key[D_SAE];           // 64 KB
    __shared__ unsigned int   hist[256];
    __shared__ unsigned long long cpack[512];
    __shared__ unsigned long long sred[256];
    __shared__ unsigned long long ap[TOPK];
    __shared__ int s_d1, s_K1, s_T, s_G, s_K2, s_nA, s_nC;

    const int tid = threadIdx.x;
    const int row = blockIdx.x;
    const float* zr = z + (size_t)row * D_SAE;
    float* fr = f + (size_t)row * D_SAE;

    // load keys (relu: non-positive -> 0), zero the f row (non-temporal:
    // f is streaming output; keep W_dec resident in L2 for the decode pass)
    for (int i = 0; i < D_SAE / 256; ++i) {
        int j = tid + i * 256;
        float v = zr[j];
        unsigned int b = __float_as_uint(v);
        key[j] = (v > 0.0f) ? (unsigned short)(b >> 15) : (unsigned short)0;
        __builtin_nontemporal_store(0.0f, &fr[j]);
    }
    if (tid == 0) { s_nA = 0; s_nC = 0; }
    hist[tid] = 0;
    __syncthreads();

    // pass 1: histogram of high 8 key bits
    for (int i = 0; i < D_SAE / 256; ++i)
        atomicAdd(&hist[key[tid + i * 256] >> 8], 1u);
    __syncthreads();
    if (tid == 0) {
        int cum = 0, d1 = 0, K1 = TOPK;
        for (int d = 255; d >= 0; --d) {
            int c = (int)hist[d];
            if (cum + c >= TOPK) { d1 = d; K1 = TOPK - cum; break; }
            cum += c;
        }
        s_d1 = d1; s_K1 = K1;
    }
    __syncthreads();
    const int d1 = s_d1, K1 = s_K1;
    hist[tid] = 0;
    __syncthreads();

    // pass 2: low 8 bits among keys with high bits == d1
    for (int i = 0; i < D_SAE / 256; ++i) {
        int k = key[tid + i * 256];
        if ((k >> 8) == d1) atomicAdd(&hist[k & 255], 1u);
    }
    __syncthreads();
    if (tid == 0) {
        int cum = 0, d2 = 0, K2 = K1;
        for (int d = 255; d >= 0; --d) {
            int c = (int)hist[d];
            if (cum + c >= K1) { d2 = d; K2 = K1 - cum; break; }
            cum += c;
        }
        s_T = (d1 << 8) | d2; s_K2 = K2; s_G = TOPK - K2;
    }
    __syncthreads();
    const int T = s_T, K2 = s_K2, G = s_G;

    // gather winners (key > T) and tie candidates (key == T)
    // pack = (valbits << 16) | (0xFFFF - idx): larger pack == larger val,
    // ties broken toward smaller index; makes final output deterministic.
    for (int i = 0; i < D_SAE / 256; ++i) {
        int j = tid + i * 256;
        int k = key[j];
        if (k > T) {
            int p = atomicAdd(&s_nA, 1);
            unsigned int vb = __float_as_uint(zr[j]);
            ap[p] = ((unsigned long long)vb << 16) | (unsigned long long)(0xFFFF - j);
        } else if (k == T) {
            int q = atomicAdd(&s_nC, 1);
            if (q < 512) {
                unsigned int vb = __float_as_uint(zr[j]);
                cpack[q] = ((unsigned long long)vb << 16) | (unsigned long long)(0xFFFF - j);
            }
        }
    }
    __syncthreads();
    const int E = (s_nC < 512) ? s_nC : 512;

    // take the K2 largest tie candidates via repeated parallel argmax
    for (int t = 0; t < K2; ++t) {
        unsigned long long m = 0;
        for (int j = tid; j < E; j += 256) { unsigned long long c = cpack[j]; if (c > m) m = c; }
        sred[tid] = m;
        __syncthreads();
        for (int s = 128; s > 0; s >>= 1) {
            if (tid < s) { unsigned long long o = sred[tid + s]; if (o > sred[tid]) sred[tid] = o; }
            __syncthreads();
        }
        unsigned long long best = sred[0];
        if (tid == 0) ap[G + t] = best;
        for (int j = tid; j < E; j += 256) if (cpack[j] == best) cpack[j] = 0;
        __syncthreads();
    }

    // bitonic sort of the 64 packs, descending
    for (unsigned size = 2; size <= 64; size <<= 1) {
        for (unsigned stride = size >> 1; stride >= 1; stride >>= 1) {
            __syncthreads();
            if (tid < 64) {
                unsigned partner = (unsigned)tid ^ stride;
                if (partner > (unsigned)tid) {
                    bool desc = ((tid & size) == 0);
                    unsigned long long a = ap[tid], b = ap[partner];
                    bool sw = desc ? (a < b) : (a > b);
                    if (sw) { ap[tid] = b; ap[partner] = a; }
                }
            }
        }
    }
    __syncthreads();

    if (tid < TOPK) {
        unsigned long long p = ap[tid];
        int j = 0xFFFF - (int)(p & 0xFFFF);
        float v = __uint_as_float((unsigned int)(p >> 16));
        fr[j] = v;
        idx_out[(size_t)row * TOPK + tid] = (float)j;
    }
}

// --------------------------------------------------------------------------
// Sparse decode: x_hat[row] = b_dec + sum_k f[row,idx_k] * W_dec[idx_k, :]
// W_dec (128 MB) is L2-resident (192 MB L2); coalesced row gathers.
// --------------------------------------------------------------------------
__global__ __launch_bounds__(256) void sae_decode(const float* __restrict__ W_dec,
                                                  const float* __restrict__ b_dec,
                                                  const float* __restrict__ f,
                                                  const float* __restrict__ idx_in,
                                                  float* __restrict__ xhat)
{
    __shared__ float sval[TOPK];
    __shared__ int   sidx[TOPK];
    const int tid = threadIdx.x;
    const int row = blockIdx.x;

    if (tid < TOPK) {
        int id = (int)idx_in[(size_t)row * TOPK + tid];
        sidx[tid] = id;
        sval[tid] = f[(size_t)row * D_SAE + id];
    }
    __syncthreads();

    float acc[4];
    #pragma unroll
    for (int j = 0; j < 4; ++j) acc[j] = b_dec[tid + j * 256];
    for (int k = 0; k < TOPK; ++k) {
        const float v = sval[k];
        const float* wr = W_dec + (size_t)sidx[k] * D_MODEL;
        #pragma unroll
        for (int j = 0; j < 4; ++j) acc[j] += v * wr[tid + j * 256];
    }
    #pragma unroll
    for (int j = 0; j < 4; ++j)
        __builtin_nontemporal_store(acc[j], &xhat[(size_t)row * D_MODEL + tid + j * 256]);
}

// --------------------------------------------------------------------------
extern "C" void kernel_launch(void* const* d_in, const int* in_sizes, int n_in,
                              void* d_out, int out_size, void* d_ws, size_t ws_size,
                              hipStream_t stream)
{
    (void)in_sizes; (void)n_in; (void)out_size; (void)d_ws; (void)ws_size;

    const float* x     = (const float*)d_in[0];
    const float* W_enc = (const float*)d_in[1];
    const float* b_enc = (const float*)d_in[2];
    const float* W_dec = (const float*)d_in[3];
    const float* b_dec = (const float*)d_in[4];

    float* out      = (float*)d_out;
    float* out_xhat = out;                                   // [B, D_MODEL]
    float* out_f    = out_xhat + (size_t)BATCH * D_MODEL;    // [B, D_SAE]
    float* out_idx  = out_f    + (size_t)BATCH * D_SAE;      // [B, K] (as float)
    float* out_z    = out_idx  + (size_t)BATCH * TOPK;       // [B, D_SAE]

    dim3 gEnc(D_SAE / 128, BATCH / 64);
    sae_encode<<<gEnc, 256, 0, stream>>>(x, W_enc, b_enc, b_dec, out_z);
    sae_topk<<<BATCH, 256, 0, stream>>>(out_z, out_f, out_idx);
    sae_decode<<<BATCH, 256, 0, stream>>>(W_dec, b_dec, out_f, out_idx, out_xhat);
}